// KVCacheAttention_6975026888714
// MI455X (gfx1250) — compile-verified
//
#include <hip/hip_runtime.h>
#include <hip/hip_bf16.h>
#include <cstddef>

#define HID    4096
#define NHEAD  32
#define HD     128
#define BQ     8       // S
#define BB     8       // B
#define BSROWS 64      // B*S
#define QKVC   12288   // 3*HID
#define KDIM   4096
#define MAXT_C 4096

typedef __attribute__((ext_vector_type(2))) float v2f;
typedef __attribute__((ext_vector_type(8))) float v8f;

static __device__ __forceinline__ v8f wmma4(v2f a, v2f b, v8f c) {
  // D = A(16x4 f32) * B(4x16 f32) + C(16x16 f32)
  return __builtin_amdgcn_wmma_f32_16x16x4_f32(false, a, false, b, (short)0, c, false, false);
}
static __device__ __forceinline__ v2f ld2(const float* p) { return *(const v2f*)p; }
static __device__ __forceinline__ v2f ld2nt(const float* p) {
  return __builtin_nontemporal_load((const v2f*)p);
}

// ---------------- LayerNorm: one block per token row ----------------
__global__ __launch_bounds__(256) void ln_kernel(const float* __restrict__ tokens,
                                                 const float* __restrict__ gamma,
                                                 const float* __restrict__ beta,
                                                 float* __restrict__ xout) {
  __shared__ float red[2][8];
  const int row = blockIdx.x;
  const float* t = tokens + (size_t)row * HID;
  float lsum = 0.f, lsq = 0.f;
  for (int i = threadIdx.x; i < HID; i += blockDim.x) {
    float v = t[i]; lsum += v; lsq += v * v;
  }
  #pragma unroll
  for (int o = 16; o; o >>= 1) { lsum += __shfl_xor(lsum, o); lsq += __shfl_xor(lsq, o); }
  const int wave = threadIdx.x >> 5, lane = threadIdx.x & 31;
  if (lane == 0) { red[0][wave] = lsum; red[1][wave] = lsq; }
  __syncthreads();
  if (threadIdx.x == 0) {
    float s = 0.f, q = 0.f;
    for (int i = 0; i < 8; ++i) { s += red[0][i]; q += red[1][i]; }
    red[0][0] = s; red[1][0] = q;
  }
  __syncthreads();
  const float mean = red[0][0] * (1.0f / HID);
  const float var  = red[1][0] * (1.0f / HID) - mean * mean;
  const float inv  = rsqrtf(var + 1e-5f);
  for (int i = threadIdx.x; i < HID; i += blockDim.x)
    xout[(size_t)row * HID + i] = (t[i] - mean) * inv * gamma[i] + beta[i];
}

// ------- GEMM: out(64 x N) = A(64 x 4096) @ W(N x 4096)^T + bias -------
// One block per 16-wide N tile; 8 waves split K; LDS tree reduction.
__global__ __launch_bounds__(256) void gemm64_kernel(const float* __restrict__ A,
                                                     const float* __restrict__ W,
                                                     const float* __restrict__ bias,
                                                     float* __restrict__ out, int N) {
  __shared__ float racc[8][4][32][8];   // 32 KB
  const int lane  = threadIdx.x & 31;
  const int wave  = threadIdx.x >> 5;
  const int ntile = blockIdx.x;
  const int n     = ntile * 16 + (lane & 15);
  const int koff  = (lane >> 4) * 2;    // K sub-offset per WMMA layout
  const int m     = lane & 15;
  const int kbeg  = wave * (KDIM / 8);
  const int kend  = kbeg + (KDIM / 8);
  v8f acc0 = {}, acc1 = {}, acc2 = {}, acc3 = {};
  const float* wp  = W + (size_t)n * KDIM + koff;
  const float* a0p = A + (size_t)(m +  0) * KDIM + koff;
  const float* a1p = A + (size_t)(m + 16) * KDIM + koff;
  const float* a2p = A + (size_t)(m + 32) * KDIM + koff;
  const float* a3p = A + (size_t)(m + 48) * KDIM + koff;
  for (int k0 = kbeg; k0 < kend; k0 += 4) {
    v2f bv = ld2nt(wp + k0);            // weights: one-touch stream -> NT
    acc0 = wmma4(ld2(a0p + k0), bv, acc0);
    acc1 = wmma4(ld2(a1p + k0), bv, acc1);
    acc2 = wmma4(ld2(a2p + k0), bv, acc2);
    acc3 = wmma4(ld2(a3p + k0), bv, acc3);
  }
  #pragma unroll
  for (int r = 0; r < 8; ++r) {
    racc[wave][0][lane][r] = acc0[r];
    racc[wave][1][lane][r] = acc1[r];
    racc[wave][2][lane][r] = acc2[r];
    racc[wave][3][lane][r] = acc3[r];
  }
  __syncthreads();
  if (wave < 4) {                        // wave-uniform
    const int mt = wave;
    float s[8];
    #pragma unroll
    for (int r = 0; r < 8; ++r) s[r] = 0.f;
    #pragma unroll
    for (int w = 0; w < 8; ++w)
      #pragma unroll
      for (int r = 0; r < 8; ++r) s[r] += racc[w][mt][lane][r];
    const float bv = bias[n];
    const int rbase = (lane >> 4) * 8;
    #pragma unroll
    for (int r = 0; r < 8; ++r) {
      const int row = mt * 16 + rbase + r;     // C layout: M = r + 8*(lane>=16)
      out[(size_t)row * N + ntile * 16 + (lane & 15)] = s[r] + bv;
    }
  }
}

// ---------------- Attention: one block per (b, h) ----------------
// Full 8 x 4096 score matrix kept in LDS (CDNA5: 320KB/WGP).
__global__ __launch_bounds__(512) void attn_kernel(const float* __restrict__ qkv,
                                                   const float* __restrict__ cache_k,
                                                   const float* __restrict__ cache_v,
                                                   const int* __restrict__ p_start,
                                                   float* __restrict__ attn_out, int maxT) {
  extern __shared__ float smem[];
  float* sc   = smem;                    // [BQ][MAXT_C] scores, then P
  float* qs   = sc + BQ * MAXT_C;        // [16][HD] (rows >= 8 zeroed)
  float* rsum = qs + 16 * HD;            // [BQ] 1/rowsum
  float* red  = rsum + BQ;               // [8][32][8] P@V partial reduction

  const int bIdx  = blockIdx.x >> 5;
  const int h     = blockIdx.x & (NHEAD - 1);
  const int lane  = threadIdx.x & 31;
  const int wave  = threadIdx.x >> 5;
  const int nwv   = blockDim.x >> 5;
  const int start = *p_start;
  const int T     = start + BQ;
  const int nt16  = (T + 15) >> 4;
  const int Tpad  = nt16 << 4;
  const float scale = rsqrtf((float)HD);

  for (int i = threadIdx.x; i < 16 * HD; i += blockDim.x) {
    const int s = i >> 7, d = i & (HD - 1);
    qs[i] = (s < BQ) ? qkv[(size_t)(bIdx * BQ + s) * QKVC + h * HD + d] * scale : 0.f;
  }
  __syncthreads();

  const int koff = (lane >> 4) * 2;
  const int m    = lane & 15;

  // ---- scores[s][t] = (q*scale) . k[t]  via WMMA over D=128 ----
  for (int tt = wave; tt < nt16; tt += nwv) {
    const int t0 = tt << 4;
    const int t  = t0 + (lane & 15);
    const int te = (t < T - 1) ? t : (T - 1);   // clamp: masked below
    const float* ksrc = (te < start)
        ? (cache_k + (((size_t)bIdx * maxT + te) * NHEAD + h) * HD)
        : (qkv + (size_t)(bIdx * BQ + (te - start)) * QKVC + HID + h * HD);
    const float* qrow = qs + m * HD + koff;
    v8f acc = {};
    #pragma unroll 8
    for (int k0 = 0; k0 < HD; k0 += 4)
      acc = wmma4(ld2(qrow + k0), ld2nt(ksrc + koff + k0), acc);
    if (lane < 16) {                     // rows 0..7 (valid); lanes 16-31 hold rows 8..15
      #pragma unroll
      for (int r = 0; r < BQ; ++r)
        sc[r * MAXT_C + t0 + lane] = (t < T) ? acc[r] : -3.0e38f;
    }
  }
  __syncthreads();

  // ---- exact softmax, wave r owns row r; keep 1/sum for later ----
  if (wave < BQ) {
    const int r = wave;
    float mx = -3.0e38f;
    for (int t = lane; t < Tpad; t += 32) mx = fmaxf(mx, sc[r * MAXT_C + t]);
    #pragma unroll
    for (int o = 16; o; o >>= 1) mx = fmaxf(mx, __shfl_xor(mx, o));
    float sum = 0.f;
    for (int t = lane; t < Tpad; t += 32) {
      const float p = __expf(sc[r * MAXT_C + t] - mx);
      sc[r * MAXT_C + t] = p;
      sum += p;
    }
    #pragma unroll
    for (int o = 16; o; o >>= 1) sum += __shfl_xor(sum, o);
    if (lane == 0) rsum[r] = 1.0f / sum;
  }
  __syncthreads();

  // ---- out[s][d] = sum_t P[s][t] * V[t][d] ; 8 d-tiles x 2 K-halves ----
  const int ntile = wave & 7;
  const int khalf = wave >> 3;
  const int d     = ntile * 16 + (lane & 15);
  const int nch   = Tpad >> 2;
  const int cbeg  = khalf * (nch >> 1);
  const int cend  = cbeg + (nch >> 1);
  const float amask = (m < BQ) ? 1.f : 0.f;   // zero phantom rows 8..15
  v8f acc = {};
  for (int c = cbeg; c < cend; ++c) {
    const int k0 = c << 2;
    v2f a = ld2(sc + (m & (BQ - 1)) * MAXT_C + k0 + koff);
    a[0] *= amask; a[1] *= amask;
    int ta = k0 + koff;     if (ta > T - 1) ta = T - 1;   // P==0 there anyway
    int tb = k0 + koff + 1; if (tb > T - 1) tb = T - 1;
    v2f bvec;
    bvec[0] = (ta < start)
        ? __builtin_nontemporal_load(cache_v + (((size_t)bIdx * maxT + ta) * NHEAD + h) * HD + d)
        : qkv[(size_t)(bIdx * BQ + (ta - start)) * QKVC + 2 * HID + h * HD + d];
    bvec[1] = (tb < start)
        ? __builtin_nontemporal_load(cache_v + (((size_t)bIdx * maxT + tb) * NHEAD + h) * HD + d)
        : qkv[(size_t)(bIdx * BQ + (tb - start)) * QKVC + 2 * HID + h * HD + d];
    acc = wmma4(a, bvec, acc);
  }
  float* myred = red + ((size_t)ntile * 32 + lane) * 8;
  if (khalf == 1) {
    #pragma unroll
    for (int r = 0; r < 8; ++r) myred[r] = acc[r];
  }
  __syncthreads();
  if (khalf == 0) {
    const int rbase = (lane >> 4) * 8;
    #pragma unroll
    for (int r = 0; r < 8; ++r) {
      const float v = acc[r] + myred[r];
      const int s = rbase + r;
      if (s < BQ)
        attn_out[(size_t)(bIdx * BQ + s) * HID + h * HD + d] = v * rsum[s];
    }
  }
}

extern "C" void kernel_launch(void* const* d_in, const int* in_sizes, int n_in,
                              void* d_out, int out_size, void* d_ws, size_t ws_size,
                              hipStream_t stream) {
  (void)n_in; (void)out_size; (void)ws_size;
  const float* tokens  = (const float*)d_in[0];
  const float* cache_k = (const float*)d_in[1];
  const float* cache_v = (const float*)d_in[2];
  const float* gamma   = (const float*)d_in[3];
  const float* beta    = (const float*)d_in[4];
  const float* qkv_w   = (const float*)d_in[5];
  const float* qkv_b   = (const float*)d_in[6];
  const float* proj_w  = (const float*)d_in[7];
  const float* proj_b  = (const float*)d_in[8];
  const int*   start   = (const int*)d_in[9];
  float* out = (float*)d_out;

  float* ws     = (float*)d_ws;
  float* xnorm  = ws;                                   // 64*4096
  float* qkvbuf = xnorm + (size_t)BSROWS * HID;         // 64*12288
  float* attn   = qkvbuf + (size_t)BSROWS * QKVC;       // 64*4096  (~5 MB total)

  const int maxT = in_sizes[1] / (BB * NHEAD * HD);

  ln_kernel<<<BSROWS, 256, 0, stream>>>(tokens, gamma, beta, xnorm);
  gemm64_kernel<<<QKVC / 16, 256, 0, stream>>>(xnorm, qkv_w, qkv_b, qkvbuf, QKVC);

  const size_t lds = (size_t)(BQ * MAXT_C + 16 * HD + BQ + 8 * 32 * 8) * sizeof(float); // ~144 KB
  hipFuncSetAttribute((const void*)attn_kernel,
                      hipFuncAttributeMaxDynamicSharedMemorySize, (int)lds);
  attn_kernel<<<BB * NHEAD, 512, lds, stream>>>(qkvbuf, cache_k, cache_v, start, attn, maxT);

  gemm64_kernel<<<HID / 16, 256, 0, stream>>>(attn, proj_w, proj_b, out, HID);
}